// tPSFNet_39934605918411
// MI455X (gfx1250) — compile-verified
//
#include <hip/hip_runtime.h>
#include <math.h>

// ---------------------------------------------------------------------------
// tPSFNet forward for MI455X (gfx1250, wave32).
//  - MLP layers run on v_wmma_f32_16x16x4_f32 (fp32 tensor path, matches ref).
//  - 99x99 depthwise conv is an exactly separable Gaussian -> two 1D passes.
//  - mk einsum collapsed analytically to a per-batch bilinear form.
// Workspace use: ~11.95 MB of d_ws (floats).
// ---------------------------------------------------------------------------

typedef __attribute__((ext_vector_type(2))) float v2f;
typedef __attribute__((ext_vector_type(8))) float v8f;

#define BATCH 256
#define HW    100
#define PIX   (HW*HW)          // 10000
#define KSZ   99
#define GSTRIDE 128            // padded stride for per-batch gaussian profile

// workspace layout (in floats)
#define WS_H1   0              // 256*256
#define WS_H2   65536          // 256*1024
#define WS_H3   327680         // 256*256
#define WS_AB   393216         // 256*4  (a, b_psf, c_mk, pad)
#define WS_G    394240         // 256*128
#define WS_TMP  427008         // 256*100*100
#define WS_DMAX 2987008        // 256
#define WS_MVAL 2987264        // 256

// output layout (in floats)
#define OUT_HR  0              // 256*1*100*100 = 2,560,000
#define OUT_LR  2560000        // 256*1*4*4     = 4,096
#define OUT_PSF 2564096        // 256*1*99*99   = 2,509,056
#define OUT_AB  5073152        // 256*1*3       = 768

__device__ __forceinline__ float softplus_f(float x) {
    return x > 20.0f ? x : log1pf(expf(x));
}

// ---------------------------------------------------------------------------
// GEMM: Y[M,N] = act( X[M,K] @ W[N,K]^T + bias[N] )
// One wave32 per 16x16 output tile; K consumed 4 at a time by
// V_WMMA_F32_16X16X4_F32.  ISA 7.12.2 layouts:
//   A 16x4 : lane(0-15)=rows, VGPR0 holds K={0|2}, VGPR1 K={1|3} per half-wave
//   B 4x16 : lane(0-15)=cols, same K split per half-wave
//   C 16x16: VGPR v = row v (lanes 0-15) / row v+8 (lanes 16-31)
// act: 0 = relu, 1 = softplus
// ---------------------------------------------------------------------------
__global__ void gemm_bias_act(const float* __restrict__ X,
                              const float* __restrict__ W,
                              const float* __restrict__ bias,
                              float* __restrict__ Y,
                              int M, int N, int K, int ldY, int act) {
    const int ntiles = (N + 15) >> 4;
    const int tm = blockIdx.x / ntiles;
    const int tn = blockIdx.x % ntiles;
    const int m0 = tm << 4;
    const int n0 = tn << 4;

    const int lane = threadIdx.x;      // 0..31, block is one wave
    const int half = lane >> 4;        // 0: K pair {0,1}, 1: K pair {2,3}
    const int l16  = lane & 15;

    const int nc   = n0 + l16;
    const bool nok = nc < N;
    const int ncc  = nok ? nc : (N - 1);
    const float wm = nok ? 1.0f : 0.0f;

    const float* xr = X + (m0 + l16) * K;
    const float* wr = W + ncc * K;

    v8f acc = {};
    for (int k = 0; k < K; k += 4) {
        const int kk = k + (half << 1);
        v2f a, b;
        a.x = xr[kk];
        a.y = xr[kk + 1];
        b.x = wr[kk] * wm;
        b.y = wr[kk + 1] * wm;
        acc = __builtin_amdgcn_wmma_f32_16x16x4_f32(
            /*neg_a=*/false, a, /*neg_b=*/false, b,
            /*c_mod=*/(short)0, acc, /*reuse_a=*/false, /*reuse_b=*/false);
    }

    if (nok) {
        const float bv = bias[nc];
        #pragma unroll
        for (int v = 0; v < 8; ++v) {
            float r = acc[v] + bv;
            r = (act == 0) ? fmaxf(r, 0.0f) : softplus_f(r);
            Y[(m0 + v + (half << 3)) * ldY + nc] = r;
        }
    }
}

// ---------------------------------------------------------------------------
// Per-batch: 1D gaussian profile g[k] = exp(-alpha*(k-49)^2) with
// alpha = (100/4802)/ab1^2   (psf_sdf = 10*d/(49*sqrt(2)))
// Also copies ab -> output.
// ---------------------------------------------------------------------------
__global__ void gauss_ab_kernel(const float* __restrict__ ws_ab,
                                float* __restrict__ g,
                                float* __restrict__ out_ab) {
    const int b = blockIdx.x;
    const int t = threadIdx.x;
    const float b1 = ws_ab[b * 4 + 1];
    const float alpha = (100.0f / 4802.0f) / (b1 * b1);
    if (t < KSZ) {
        const float d = (float)t - 49.0f;
        g[b * GSTRIDE + t] = expf(-alpha * d * d);
    }
    if (t < 3) out_ab[b * 3 + t] = ws_ab[b * 4 + t];
}

// psf[b,i,j] = a_b * g[b,i] * g[b,j]
__global__ void psf_kernel(const float* __restrict__ ws_ab,
                           const float* __restrict__ g,
                           float* __restrict__ psf) {
    const int idx = blockIdx.x * blockDim.x + threadIdx.x;
    if (idx >= BATCH * KSZ * KSZ) return;
    const int b = idx / (KSZ * KSZ);
    const int r = idx % (KSZ * KSZ);
    const int i = r / KSZ;
    const int j = r % KSZ;
    psf[idx] = ws_ab[b * 4] * g[b * GSTRIDE + i] * g[b * GSTRIDE + j];
}

// per-batch max of depth
__global__ void batch_max_kernel(const float* __restrict__ depth,
                                 float* __restrict__ dmax) {
    const int b = blockIdx.x;
    const float* p = depth + b * PIX;
    float m = -INFINITY;
    for (int i = threadIdx.x; i < PIX; i += 256) m = fmaxf(m, p[i]);
    __shared__ float s[256];
    s[threadIdx.x] = m;
    __syncthreads();
    for (int o = 128; o > 0; o >>= 1) {
        if (threadIdx.x < o) s[threadIdx.x] = fmaxf(s[threadIdx.x], s[threadIdx.x + o]);
        __syncthreads();
    }
    if (threadIdx.x == 0) dmax[b] = s[0];
}

// horizontal 99-tap pass: tmp[b,r,c] = sum_k depth[b,r,c+k-49]*g[b,k]
__global__ void conv_h_kernel(const float* __restrict__ depth,
                              const float* __restrict__ g,
                              float* __restrict__ tmp) {
    const int r = blockIdx.x;
    const int b = blockIdx.y;
    __shared__ float row[HW];
    __shared__ float gk[KSZ];
    const int t = threadIdx.x;
    if (t < HW)  row[t] = depth[(b * HW + r) * HW + t];
    if (t < KSZ) gk[t]  = g[b * GSTRIDE + t];
    __syncthreads();
    if (t < HW) {
        const int k0 = max(0, 49 - t);
        const int k1 = min(KSZ - 1, 148 - t);
        float s = 0.0f;
        for (int k = k0; k <= k1; ++k) s = fmaf(row[t + k - 49], gk[k], s);
        tmp[(b * HW + r) * HW + t] = s;
    }
}

// vertical pass + amplitude: hr[b,oi,oj] = a_b * sum_k tmp[b,oi+k-49,oj]*g[b,k]
__global__ void conv_v_kernel(const float* __restrict__ tmp,
                              const float* __restrict__ g,
                              const float* __restrict__ ws_ab,
                              float* __restrict__ hr) {
    const int oi = blockIdx.x;
    const int b  = blockIdx.y;
    __shared__ float gk[KSZ];
    const int t = threadIdx.x;
    if (t < KSZ) gk[t] = g[b * GSTRIDE + t];
    __syncthreads();
    if (t < HW) {
        const int k0 = max(0, 49 - oi);
        const int k1 = min(KSZ - 1, 148 - oi);
        float s = 0.0f;
        for (int k = k0; k <= k1; ++k)
            s = fmaf(tmp[(b * HW + oi + k - 49) * HW + t], gk[k], s);
        hr[(b * HW + oi) * HW + t] = ws_ab[b * 4] * s;
    }
}

// mval[b] = max(0, max over !mask of hr)   (mask = depth > dmax-1e-3)
__global__ void mval_kernel(const float* __restrict__ depth,
                            const float* __restrict__ dmax,
                            const float* __restrict__ hr,
                            float* __restrict__ mval) {
    const int b = blockIdx.x;
    const float thr = dmax[b] - 0.001f;
    float m = 0.0f;
    for (int i = threadIdx.x; i < PIX; i += 256) {
        const float d = depth[b * PIX + i];
        if (!(d > thr)) m = fmaxf(m, hr[b * PIX + i]);
    }
    __shared__ float s[256];
    s[threadIdx.x] = m;
    __syncthreads();
    for (int o = 128; o > 0; o >>= 1) {
        if (threadIdx.x < o) s[threadIdx.x] = fmaxf(s[threadIdx.x], s[threadIdx.x + o]);
        __syncthreads();
    }
    if (threadIdx.x == 0) mval[b] = s[0];
}

// hr = where(mask, mval, hr)  (in place on d_out)
__global__ void finalize_hr_kernel(const float* __restrict__ depth,
                                   const float* __restrict__ dmax,
                                   const float* __restrict__ mval,
                                   float* __restrict__ hr) {
    const int idx = blockIdx.x * blockDim.x + threadIdx.x;
    if (idx >= BATCH * PIX) return;
    const int b = idx / PIX;
    if (depth[idx] > dmax[b] - 0.001f) hr[idx] = mval[b];
}

// lr[b,i,j] = 1e-4 * (u_i^T * hr_b * u_j - mn*sum(hr_b)) / (1 - mn)
// u_i(h) = exp(-gamma*(h-(12+25i))^2), gamma=(100/15138)/ab2, mn=exp(-100/ab2)
__global__ void lr_kernel(const float* __restrict__ hr,
                          const float* __restrict__ ws_ab,
                          float* __restrict__ lr) {
    const int b = blockIdx.x;
    const int t = threadIdx.x;          // 128 threads
    const float ab2 = ws_ab[b * 4 + 2];
    const float gamma = (100.0f / 15138.0f) / ab2;

    __shared__ float u[4][HW];
    __shared__ float q[4][HW];
    __shared__ float red[128];

    for (int idx = t; idx < 4 * HW; idx += 128) {
        const int i = idx / HW, h = idx % HW;
        const float dx = (float)h - (12.0f + 25.0f * (float)i);
        u[i][h] = expf(-gamma * dx * dx);
    }
    __syncthreads();

    // q[j][h] = sum_w hr[b,h,w] * u[j][w]
    for (int idx = t; idx < 4 * HW; idx += 128) {
        const int j = idx / HW, h = idx % HW;
        const float* hrow = hr + (b * HW + h) * HW;
        float s = 0.0f;
        for (int w = 0; w < HW; ++w) s = fmaf(hrow[w], u[j][w], s);
        q[j][h] = s;
    }

    // T = sum of hr_b
    float ts = 0.0f;
    for (int i = t; i < PIX; i += 128) ts += hr[b * PIX + i];
    red[t] = ts;
    __syncthreads();
    for (int o = 64; o > 0; o >>= 1) {
        if (t < o) red[t] += red[t + o];
        __syncthreads();
    }

    if (t < 16) {
        const int i = t >> 2, j = t & 3;
        float s = 0.0f;
        for (int h = 0; h < HW; ++h) s = fmaf(u[i][h], q[j][h], s);
        const float mn = expf(-100.0f / ab2);
        lr[b * 16 + t] = 1e-4f * (s - mn * red[0]) / (1.0f - mn);
    }
}

// ---------------------------------------------------------------------------
extern "C" void kernel_launch(void* const* d_in, const int* in_sizes, int n_in,
                              void* d_out, int out_size, void* d_ws, size_t ws_size,
                              hipStream_t stream) {
    const float* x     = (const float*)d_in[0];   // 256x3x4x4 = 256x48
    const float* depth = (const float*)d_in[1];   // 256x1x100x100
    const float* W1    = (const float*)d_in[2];
    const float* b1    = (const float*)d_in[3];
    const float* W2    = (const float*)d_in[4];
    const float* b2    = (const float*)d_in[5];
    const float* W3    = (const float*)d_in[6];
    const float* b3    = (const float*)d_in[7];
    const float* W4    = (const float*)d_in[8];
    const float* b4    = (const float*)d_in[9];

    float* out = (float*)d_out;
    float* ws  = (float*)d_ws;

    float* h1   = ws + WS_H1;
    float* h2   = ws + WS_H2;
    float* h3   = ws + WS_H3;
    float* wsab = ws + WS_AB;
    float* g    = ws + WS_G;
    float* tmp  = ws + WS_TMP;
    float* dmax = ws + WS_DMAX;
    float* mval = ws + WS_MVAL;

    float* out_hr  = out + OUT_HR;
    float* out_lr  = out + OUT_LR;
    float* out_psf = out + OUT_PSF;
    float* out_ab  = out + OUT_AB;

    // MLP: 48 -> 256 -> 1024 -> 256 -> 3  (WMMA fp32 tiles)
    gemm_bias_act<<<16 * 16,  32, 0, stream>>>(x,  W1, b1, h1,  BATCH,  256,   48,  256, 0);
    gemm_bias_act<<<16 * 64,  32, 0, stream>>>(h1, W2, b2, h2,  BATCH, 1024,  256, 1024, 0);
    gemm_bias_act<<<16 * 16,  32, 0, stream>>>(h2, W3, b3, h3,  BATCH,  256, 1024,  256, 0);
    gemm_bias_act<<<16 *  1,  32, 0, stream>>>(h3, W4, b4, wsab, BATCH,   3,  256,    4, 1);

    // Gaussian profiles + ab output
    gauss_ab_kernel<<<BATCH, 128, 0, stream>>>(wsab, g, out_ab);

    // psf output (separable product)
    psf_kernel<<<(BATCH * KSZ * KSZ) / 256, 256, 0, stream>>>(wsab, g, out_psf);

    // per-batch depth max
    batch_max_kernel<<<BATCH, 256, 0, stream>>>(depth, dmax);

    // separable 99-tap depthwise conv
    conv_h_kernel<<<dim3(HW, BATCH), 128, 0, stream>>>(depth, g, tmp);
    conv_v_kernel<<<dim3(HW, BATCH), 128, 0, stream>>>(tmp, g, wsab, out_hr);

    // masked-max replacement
    mval_kernel<<<BATCH, 256, 0, stream>>>(depth, dmax, out_hr, mval);
    finalize_hr_kernel<<<(BATCH * PIX) / 256, 256, 0, stream>>>(depth, dmax, mval, out_hr);

    // lr bilinear form
    lr_kernel<<<BATCH, 128, 0, stream>>>(out_hr, wsab, out_lr);
}